// LinearEncoder_66958540144842
// MI455X (gfx1250) — compile-verified
//
#include <hip/hip_runtime.h>
#include <hip/hip_bf16.h>

typedef __attribute__((ext_vector_type(2))) float v2f;
typedef __attribute__((ext_vector_type(8))) float v8f;

#define COUT 16

// ---------------------------------------------------------------- deg init
__global__ void gcn_deg_init(float* __restrict__ deg, int n) {
    int i = blockIdx.x * blockDim.x + threadIdx.x;
    if (i < n) deg[i] = 1.0f;  // self-loop weight
}

// ------------------------------------------------- deg += w over dst edges
__global__ void gcn_deg_edges(float* __restrict__ deg,
                              const long long* __restrict__ ei,  // [2,E] row-major
                              const float* __restrict__ ew, int e) {
    int i = blockIdx.x * blockDim.x + threadIdx.x;
    if (i < e) {
        long long d = ei[(long long)e + i];  // dst row
        atomicAdd(&deg[d], ew[i]);
    }
}

// --------------------------------------------------------------- dis=rsqrt
__global__ void gcn_rsqrt(const float* __restrict__ deg, float* __restrict__ dis, int n) {
    int i = blockIdx.x * blockDim.x + threadIdx.x;
    if (i < n) {
        float dv = deg[i];
        dis[i] = dv > 0.0f ? rsqrtf(fmaxf(dv, 1e-12f)) : 0.0f;
    }
}

// --------------------------------------- h = x @ W via V_WMMA_F32_16X16X4_F32
// One wave32 owns a 16(M) x 16(N) fp32 tile; K loops 0..c_in step 4.
// f32 A 16x4 layout : lanes 0-15 -> M=lane, V0=K0 V1=K1 ; lanes 16-31 -> K2,K3
// f32 B  4x16 layout: lanes 0-15 -> N=lane, V0=K0 V1=K1 ; lanes 16-31 -> K2,K3
// f32 C/D 16x16     : VGPR r: lanes 0-15 M=r, lanes 16-31 M=r+8, N=lane&15
__global__ __launch_bounds__(256) void gcn_gemm_wmma(const float* __restrict__ x,
                                                     const float* __restrict__ W,
                                                     float* __restrict__ h,
                                                     int n_nodes, int c_in) {
    __shared__ float lW[512 * COUT];  // 32 KB, c_in <= 512
    int tid = threadIdx.x;
    int nW = c_in * COUT;
    for (int i = tid; i < nW; i += blockDim.x) lW[i] = W[i];
    __syncthreads();

    int wave = blockIdx.x * (blockDim.x >> 5) + (tid >> 5);
    int lane = tid & 31;
    int row0 = wave * 16;
    if (row0 >= n_nodes) return;  // uniform per wave -> EXEC stays all-1s

    int m     = lane & 15;          // M (for A) / N (for B)
    int khalf = (lane >> 4) << 1;   // 0 for lanes 0-15, 2 for lanes 16-31

    const float* xr = x + (size_t)(row0 + m) * c_in + khalf;
    v8f acc = {};
    for (int k = 0; k < c_in; k += 4) {
        v2f a;
        a.x = xr[k];
        a.y = xr[k + 1];
        v2f bf;
        bf.x = lW[(k + khalf) * COUT + m];
        bf.y = lW[(k + khalf + 1) * COUT + m];
        acc = __builtin_amdgcn_wmma_f32_16x16x4_f32(false, a, false, bf,
                                                    (short)0, acc, false, false);
    }

    int mout = (lane >> 4) << 3;  // 0 or 8
    float* hr = h + (size_t)(row0 + mout) * COUT + m;
#pragma unroll
    for (int r = 0; r < 8; ++r) hr[(size_t)r * COUT] = acc[r];
}

// ---------------------------- out = b + h * dis^2   (bias + fused self-loop)
__global__ void gcn_out_init(const float* __restrict__ h, const float* __restrict__ dis,
                             const float* __restrict__ b, float* __restrict__ out, int n) {
    long long t = (long long)blockIdx.x * blockDim.x + threadIdx.x;
    if (t >= (long long)n * COUT) return;
    int c = (int)(t & (COUT - 1));
    long long i = t >> 4;
    float dv = dis[i];
    out[t] = b[c] + h[t] * dv * dv;
}

// ------------------- out[dst] += dis[src]*w*dis[dst] * h[src], 16 lanes/edge
__global__ __launch_bounds__(256) void gcn_scatter(const long long* __restrict__ ei,
                                                   const float* __restrict__ ew,
                                                   const float* __restrict__ dis,
                                                   const float* __restrict__ h,
                                                   float* __restrict__ out, int e) {
    long long t = (long long)blockIdx.x * blockDim.x + threadIdx.x;
    int c = (int)(t & (COUT - 1));
    long long idx = t >> 4;
    if (idx >= e) return;
    long long s = ei[idx];
    long long d = ei[(long long)e + idx];
    float nrm = dis[s] * ew[idx] * dis[d];
    atomicAdd(&out[d * COUT + c], h[s * COUT + c] * nrm);
}

extern "C" void kernel_launch(void* const* d_in, const int* in_sizes, int n_in,
                              void* d_out, int out_size, void* d_ws, size_t ws_size,
                              hipStream_t stream) {
    const float*     x  = (const float*)d_in[0];
    const long long* ei = (const long long*)d_in[1];   // int64 [2,E]
    const float*     ew = (const float*)d_in[2];
    const float*     W  = (const float*)d_in[3];
    const float*     b  = (const float*)d_in[4];
    float* out = (float*)d_out;

    const int c_out = in_sizes[4];                 // 16
    const int c_in  = in_sizes[3] / c_out;         // 512
    const int n     = in_sizes[0] / c_in;          // 100000
    const int e     = in_sizes[2];                 // 3200000

    float* deg = (float*)d_ws;          // [n]
    float* dis = deg + n;               // [n]
    float* h   = dis + n;               // [n*16]

    const int B = 256;
    gcn_deg_init<<<(n + B - 1) / B, B, 0, stream>>>(deg, n);
    gcn_deg_edges<<<(e + B - 1) / B, B, 0, stream>>>(deg, ei, ew, e);
    gcn_rsqrt<<<(n + B - 1) / B, B, 0, stream>>>(deg, dis, n);

    int waves  = (n + 15) / 16;                    // one 16x16 tile per wave
    int blocks = (waves + (B / 32) - 1) / (B / 32);
    gcn_gemm_wmma<<<blocks, B, 0, stream>>>(x, W, h, n, c_in);

    long long tot_o = (long long)n * c_out;
    gcn_out_init<<<(int)((tot_o + B - 1) / B), B, 0, stream>>>(h, dis, b, out, n);

    long long tot_s = (long long)e * c_out;
    gcn_scatter<<<(int)((tot_s + B - 1) / B), B, 0, stream>>>(ei, ew, dis, h, out, e);
}